// MinibatchDiscrimination_82772609728905
// MI455X (gfx1250) — compile-verified
//
#include <hip/hip_runtime.h>
#include <hip/hip_bf16.h>

// ---------------------------------------------------------------------------
// MinibatchDiscrimination for MI455X (gfx1250, wave32, WMMA + async-to-LDS)
//   x: [128,1024] f32, T: [1024,512,16] f32
//   m = x @ T.reshape(1024, 8192)          -> WMMA f32 16x16x4 GEMM,
//                                             T staged via GLOBAL_LOAD_ASYNC_TO_LDS
//   o_b[b,o] = sum_a exp(-sum_k |m[a,o,k]-m[b,o,k]|) - 1   -> LDS tile kernel
//   out = concat([x, o_b], axis=1)  [128, 1536] f32
// ---------------------------------------------------------------------------

#define BATCH   128
#define INF     1024
#define OUTF    512
#define KD      16
#define NCOL    (OUTF * KD)   // 8192 flattened columns of T
#define OUTW    (INF + OUTF)  // 1536

#define KC      32            // k-rows staged per LDS buffer
#define NSTAGE  (INF / KC)    // 32 stages
#define LDSROW  80            // padded row stride (80*2 % 64 == 32 -> no bank clash)
#define LDSBUF  (KC * LDSROW) // floats per buffer (2560)

typedef __attribute__((ext_vector_type(8))) float v8f;
typedef __attribute__((ext_vector_type(2))) float v2f;
typedef __attribute__((ext_vector_type(4))) int   v4i;

typedef __attribute__((address_space(1))) v4i gv4i;  // global 16B vector
typedef __attribute__((address_space(3))) v4i lv4i;  // LDS 16B vector

#if defined(__has_builtin)
#if __has_builtin(__builtin_amdgcn_global_load_async_to_lds_b128)
#define HAVE_ASYNC_B128 1
#endif
#if __has_builtin(__builtin_amdgcn_s_wait_asynccnt)
#define HAVE_WAIT_ASYNC 1
#endif
#endif

// 16-byte global -> LDS async copy (no VGPR round trip, tracked by ASYNCcnt)
__device__ __forceinline__ void async_copy16(const float* g, float* l) {
#ifdef HAVE_ASYNC_B128
  __builtin_amdgcn_global_load_async_to_lds_b128((gv4i*)g, (lv4i*)l, 0, 0);
#else
  unsigned loff = (unsigned)(size_t)l;  // wave-relative LDS byte address
  asm volatile("global_load_async_to_lds_b128 %0, %1, off"
               :: "v"(loff), "v"(g) : "memory");
#endif
}

__device__ __forceinline__ void wait_asynccnt0() {
#ifdef HAVE_WAIT_ASYNC
  __builtin_amdgcn_s_wait_asynccnt(0);
#else
  asm volatile("s_wait_asynccnt 0x0" ::: "memory");
#endif
}

// ---------------------------------------------------------------------------
// Kernel 1: m[128, 8192] = x[128,1024] * T[1024,8192] via V_WMMA_F32_16X16X4_F32
// One block (8 waves) owns a 64-wide column strip for all 128 rows, so each
// strip of T is fetched from global exactly once (32 MB total vs 256 MB when
// every row-block re-reads it). T tiles are double-buffered in LDS via
// GLOBAL_LOAD_ASYNC_TO_LDS_B128; WMMAs of stage s overlap the copies of s+1.
// A 16x4 f32 layout: lanes 0-15 hold K=0 (v0), K=1 (v1); lanes 16-31 K=2,K=3.
// B 4x16 f32 layout (transpose of A): lane%16 = N; v0 = K=0|2, v1 = K=1|3.
// D 16x16 f32 layout: lane%16 = N; VGPR r holds M=r (lanes 0-15) / M=r+8.
// ---------------------------------------------------------------------------
__global__ __launch_bounds__(256) void mbd_gemm_wmma(
    const float* __restrict__ x, const float* __restrict__ T,
    float* __restrict__ m) {
  __shared__ float ldsT[2 * LDSBUF];  // 20 KB

  const int tid   = threadIdx.x;       // 0..255
  const int lane  = tid & 31;
  const int wv    = tid >> 5;          // wave id == row block mb (0..7)
  const int n0    = blockIdx.x << 6;   // 64-wide column strip
  const int nlane = lane & 15;         // N within tile / M row for A
  const int khalf = lane >> 4;         // 0: K=0,1   1: K=2,3

  // Async staging map: thread t moves 8 floats (two B128) of the KCx64 tile.
  const int krow = tid >> 3;               // 0..31
  const int col  = (tid & 7) << 3;         // 0,8,...,56
  const float* gsrc = T + (size_t)krow * NCOL + n0 + col;
  float*       lbase = &ldsT[krow * LDSROW + col];

  // Prologue: stage 0 into buffer 0.
  async_copy16(gsrc + 0, lbase + 0);
  async_copy16(gsrc + 4, lbase + 4);

  v8f acc0 = {}, acc1 = {}, acc2 = {}, acc3 = {};
  const float* xrow = x + (size_t)(wv * 16 + nlane) * INF;

  for (int ks = 0; ks < NSTAGE; ++ks) {
    const int buf = ks & 1;
    wait_asynccnt0();   // my copies for this buffer landed
    __syncthreads();    // everyone's copies landed

    if (ks + 1 < NSTAGE) {  // kick copies for the next stage (other buffer)
      const float* g = gsrc + (size_t)(ks + 1) * KC * NCOL;
      float*       l = lbase + (buf ^ 1) * LDSBUF;
      async_copy16(g + 0, l + 0);
      async_copy16(g + 4, l + 4);
    }

    const float* xs = xrow + ks * KC + khalf * 2;
#pragma unroll
    for (int kk = 0; kk < KC / 4; ++kk) {
      v2f a;
      a.x = xs[kk * 4 + 0];
      a.y = xs[kk * 4 + 1];

      const float* lb =
          &ldsT[buf * LDSBUF + (kk * 4 + khalf * 2) * LDSROW + nlane];
      v2f b0, b1, b2, b3;
      b0.x = lb[0];  b0.y = lb[LDSROW + 0];
      b1.x = lb[16]; b1.y = lb[LDSROW + 16];
      b2.x = lb[32]; b2.y = lb[LDSROW + 32];
      b3.x = lb[48]; b3.y = lb[LDSROW + 48];

      acc0 = __builtin_amdgcn_wmma_f32_16x16x4_f32(false, a, false, b0,
                                                   (short)0, acc0, false, false);
      acc1 = __builtin_amdgcn_wmma_f32_16x16x4_f32(false, a, false, b1,
                                                   (short)0, acc1, false, false);
      acc2 = __builtin_amdgcn_wmma_f32_16x16x4_f32(false, a, false, b2,
                                                   (short)0, acc2, false, false);
      acc3 = __builtin_amdgcn_wmma_f32_16x16x4_f32(false, a, false, b3,
                                                   (short)0, acc3, false, false);
    }
  }

  const int colbase = n0 + nlane;
  const int rowbase = wv * 16 + khalf * 8;
#pragma unroll
  for (int r = 0; r < 8; ++r) {
    float* mr = m + (size_t)(rowbase + r) * NCOL + colbase;
    mr[0]  = acc0[r];
    mr[16] = acc1[r];
    mr[32] = acc2[r];
    mr[48] = acc3[r];
  }
}

// ---------------------------------------------------------------------------
// Kernel 2: per output-feature pairwise L1 + exp reduction.
// One block of 128 threads per feature o. Stage the 128x16 f32 tile in LDS
// (stride 17 -> conflict-free writes; reads of tile[a][k] are uniform across
// the wave -> LDS broadcast). Own row stays in registers.
// ---------------------------------------------------------------------------
__global__ __launch_bounds__(128) void mbd_pairwise(
    const float* __restrict__ m, float* __restrict__ out) {
  __shared__ float tile[BATCH][KD + 1];

  const int o = blockIdx.x;    // 0..511
  const int b = threadIdx.x;   // 0..127

  float rb[KD];
  const float* mrow = m + (size_t)b * NCOL + (size_t)o * KD;
#pragma unroll
  for (int k = 0; k < KD; ++k) {
    rb[k] = mrow[k];
    tile[b][k] = rb[k];
  }
  __syncthreads();

  float acc = 0.0f;
  for (int a = 0; a < BATCH; ++a) {
    float s = 0.0f;
#pragma unroll
    for (int k = 0; k < KD; ++k) s += fabsf(tile[a][k] - rb[k]);
    acc += __expf(-s);
  }

  out[(size_t)b * OUTW + INF + o] = acc - 1.0f;  // remove self-similarity
}

// ---------------------------------------------------------------------------
// Kernel 3: copy x into out[:, :1024] (strided concat).
// ---------------------------------------------------------------------------
__global__ __launch_bounds__(256) void mbd_copy_x(
    const float* __restrict__ x, float* __restrict__ out) {
  const int t = blockIdx.x * 256 + threadIdx.x;  // 0..131071
  const int b = t >> 10;
  const int i = t & 1023;
  out[(size_t)b * OUTW + i] = x[(size_t)b * INF + i];
}

// ---------------------------------------------------------------------------
extern "C" void kernel_launch(void* const* d_in, const int* in_sizes, int n_in,
                              void* d_out, int out_size, void* d_ws,
                              size_t ws_size, hipStream_t stream) {
  const float* x = (const float*)d_in[0];   // [128, 1024]
  const float* T = (const float*)d_in[1];   // [1024, 512, 16]
  float* out     = (float*)d_out;           // [128, 1536]
  float* m_ws    = (float*)d_ws;            // [128, 8192] = 4 MB scratch

  // Phase 1: WMMA GEMM (T staged once through LDS via async copies)
  mbd_gemm_wmma<<<128, 256, 0, stream>>>(x, T, m_ws);
  // Phase 2: pairwise L1 / exp reduction -> out[:, 1024:]
  mbd_pairwise<<<OUTF, BATCH, 0, stream>>>(m_ws, out);
  // Phase 3: passthrough concat -> out[:, :1024]
  mbd_copy_x<<<512, 256, 0, stream>>>(x, out);
}